// GraphAttention_88467736363048
// MI455X (gfx1250) — compile-verified
//
#include <hip/hip_runtime.h>
#include <hip/hip_bf16.h>
#include <stdint.h>

#define ALPHA 0.2f
#define NN   8192
#define FIN  512
#define FOUT 64

typedef __attribute__((ext_vector_type(2))) float v2f;
typedef __attribute__((ext_vector_type(8))) float v8f;
typedef __attribute__((ext_vector_type(4))) int   v4i;

// ---- CDNA5 async-to-LDS path (guarded: falls back to plain load+ds_store) ----
#if defined(__has_builtin)
#  if __has_builtin(__builtin_amdgcn_global_load_async_to_lds_b128) && \
      __has_builtin(__builtin_amdgcn_s_wait_asynccnt)
#    define USE_ASYNC 1
#  endif
#endif
#ifndef USE_ASYNC
#  define USE_ASYNC 0
#endif

#if USE_ASYNC
typedef __attribute__((address_space(1))) v4i* as1_v4i_p;   // global
typedef __attribute__((address_space(3))) v4i* as3_v4i_p;   // LDS (32-bit ptr)
__device__ __forceinline__ void asyncB128(const void* g, void* l) {
  __builtin_amdgcn_global_load_async_to_lds_b128(
      (as1_v4i_p)(uintptr_t)g,
      (as3_v4i_p)(uint32_t)(uintptr_t)l, 0, 0);
}
__device__ __forceinline__ void asyncWait0() { __builtin_amdgcn_s_wait_asynccnt(0); }
#else
__device__ __forceinline__ void asyncWait0() {}
#endif

// ------------------------------------------------------------------
// Kernel A: z = input @ W  (8192x512 @ 512x64), f32 WMMA 16x16x4.
// One wave per 16-row tile; K staged through LDS (padded rows, bank-clean).
// ------------------------------------------------------------------
__global__ __launch_bounds__(32) void zgemm_kernel(const float* __restrict__ X,
                                                   const float* __restrict__ W,
                                                   float* __restrict__ Z)
{
  __shared__ float ta[16][68];            // 16 x 64 K-chunk, +4 pad (bank stride 4)
  const int lane = threadIdx.x;
  const int m  = lane & 15;
  const int kh = lane >> 4;               // 0: K even pair, 1: K odd pair (+2)
  const int i0 = blockIdx.x * 16;

  v8f acc[4] = {};

  for (int k0 = 0; k0 < FIN; k0 += 64) {
    // cooperative coalesced load of 16x64 input chunk
    #pragma unroll
    for (int t = 0; t < 8; ++t) {
      int idx = t * 32 + lane;            // 0..255 float4s
      int r = idx >> 4;
      int c = (idx & 15) << 2;
      *(float4*)&ta[r][c] = *(const float4*)&X[(size_t)(i0 + r) * FIN + k0 + c];
    }
    __syncthreads();
    #pragma unroll
    for (int kk = 0; kk < 64; kk += 4) {
      v2f A;                              // A 16x4 frag: lane m row, K = kk+2*kh, +1
      A.x = ta[m][kk + 2 * kh];
      A.y = ta[m][kk + 2 * kh + 1];
      const int kr = k0 + kk + 2 * kh;
      #pragma unroll
      for (int nt = 0; nt < 4; ++nt) {
        v2f B;                            // B 4x16 frag: K rows of W, N = nt*16+m
        B.x = W[(size_t)kr * FOUT + nt * 16 + m];
        B.y = W[(size_t)(kr + 1) * FOUT + nt * 16 + m];
        acc[nt] = __builtin_amdgcn_wmma_f32_16x16x4_f32(
            false, A, false, B, (short)0, acc[nt], false, false);
      }
    }
    __syncthreads();
  }
  // C frag: VGPR v -> row v (lanes 0-15) / v+8 (lanes 16-31), col = lane%16
  #pragma unroll
  for (int nt = 0; nt < 4; ++nt)
    #pragma unroll
    for (int v = 0; v < 8; ++v)
      Z[(size_t)(i0 + (kh << 3) + v) * FOUT + nt * 16 + m] = acc[nt][v];
}

// ------------------------------------------------------------------
// Kernel B: s1 = z@a1, s2 = z@a2  (one wave per row, shuffle reduce)
// ------------------------------------------------------------------
__global__ __launch_bounds__(256) void s12_kernel(const float* __restrict__ z,
                                                  const float* __restrict__ a,
                                                  float* __restrict__ s1,
                                                  float* __restrict__ s2)
{
  const int lane = threadIdx.x & 31;
  const int w    = threadIdx.x >> 5;
  const int r    = blockIdx.x * 8 + w;
  const float z0 = z[(size_t)r * FOUT + lane];
  const float z1 = z[(size_t)r * FOUT + lane + 32];
  float p1 = z0 * a[lane]      + z1 * a[lane + 32];
  float p2 = z0 * a[64 + lane] + z1 * a[96 + lane];
  #pragma unroll
  for (int off = 16; off >= 1; off >>= 1) {
    p1 += __shfl_xor(p1, off);
    p2 += __shfl_xor(p2, off);
  }
  if (lane == 0) { s1[r] = p1; s2[r] = p2; }
}

// ------------------------------------------------------------------
// Kernel C: S2max = max(s2)  (single block)
// ------------------------------------------------------------------
__global__ __launch_bounds__(256) void s2max_kernel(const float* __restrict__ s2,
                                                    float* __restrict__ out)
{
  __shared__ float red[256];
  const int tid = threadIdx.x;
  float mx = -3.4e38f;
  for (int t = tid; t < NN; t += 256) mx = fmaxf(mx, s2[t]);
  red[tid] = mx;
  __syncthreads();
  for (int off = 128; off > 0; off >>= 1) {
    if (tid < off) red[tid] = fmaxf(red[tid], red[tid + off]);
    __syncthreads();
  }
  if (tid == 0) out[0] = red[0];
}

// ------------------------------------------------------------------
// Kernel D: fused masked-softmax attention @ z, single pass over adj.
// Block = 8 waves; block owns 128 rows x 2048-wide j slice.
// Wave w owns rows i0+w*16 .. +15; all waves share the z tile stream.
// Double-buffered LDS tiles; async-to-LDS when available.
// ------------------------------------------------------------------
__global__ __launch_bounds__(256) void attn_kernel(
    const int*   __restrict__ adj,  const float* __restrict__ z,
    const float* __restrict__ s1,   const float* __restrict__ s2,
    const float* __restrict__ s2max_p,
    float* __restrict__ accP, float* __restrict__ denomP)
{
  constexpr int TJ = 32;
  constexpr int NT = 2048 / TJ;                 // 64 tiles per block

  __shared__ float zt[2][TJ][80];               // pad 80: halves hit disjoint banks
  __shared__ int   adjt[8][2][16][36];          // pad 36: 16 rows -> 16 banks
  __shared__ float s2t[2][TJ];

  const int tid  = threadIdx.x;
  const int lane = tid & 31;
  const int w    = tid >> 5;
  const int m    = lane & 15;
  const int kh   = lane >> 4;
  const int i0   = blockIdx.x * 128 + w * 16;
  const int jbase = blockIdx.y * 2048;

  const float S2max = s2max_p[0];
  const float s1m   = s1[i0 + m];
  const float tmx   = s1m + S2max;
  const float cmax  = tmx > 0.f ? tmx : ALPHA * tmx;   // per-row softmax shift bound

  v8f acc[4] = {};
  float dpart = 0.f;

  auto load_tile = [&](int buf, int j0) {
    // shared z tile: 32x64 f32 = 512 x b128, 2 per thread (coalesced 256B rows)
    #pragma unroll
    for (int t = 0; t < 2; ++t) {
      int idx = t * 256 + tid;
      int r = idx >> 4;
      int c = (idx & 15) << 2;
#if USE_ASYNC
      asyncB128(&z[(size_t)(j0 + r) * FOUT + c], &zt[buf][r][c]);
#else
      *(float4*)&zt[buf][r][c] = *(const float4*)&z[(size_t)(j0 + r) * FOUT + c];
#endif
    }
    // per-wave adj tile: 16x32 i32 = 128 x b128, 4 per lane (128B segments)
    #pragma unroll
    for (int t = 0; t < 4; ++t) {
      int idx = t * 32 + lane;
      int r = idx >> 3;
      int c = (idx & 7) << 2;
#if USE_ASYNC
      asyncB128(&adj[(size_t)(i0 + r) * NN + j0 + c], &adjt[w][buf][r][c]);
#else
      *(int4*)&adjt[w][buf][r][c] = *(const int4*)&adj[(size_t)(i0 + r) * NN + j0 + c];
#endif
    }
    if (tid < TJ) s2t[buf][tid] = s2[j0 + tid];
  };

  load_tile(0, jbase);
  asyncWait0();
  __syncthreads();

  for (int t = 0; t < NT; ++t) {
    const int buf = t & 1;
    if (t + 1 < NT) load_tile(buf ^ 1, jbase + (t + 1) * TJ);   // prefetch overlaps compute

    #pragma unroll
    for (int kk = 0; kk < TJ; kk += 4) {
      const int ja = kk + (kh << 1);                 // this lane's K (=j) pair
      const int2 av = *(const int2*)&adjt[w][buf][m][ja];
      float e0 = s1m + s2t[buf][ja];
      float e1 = s1m + s2t[buf][ja + 1];
      e0 = (e0 > 0.f ? e0 : ALPHA * e0) - cmax;
      e1 = (e1 > 0.f ? e1 : ALPHA * e1) - cmax;
      const float p0 = (av.x > 0) ? __expf(e0) : 0.f;
      const float p1 = (av.y > 0) ? __expf(e1) : 0.f;
      dpart += p0 + p1;
      v2f A; A.x = p0; A.y = p1;                     // P in A-frag layout
      #pragma unroll
      for (int nt = 0; nt < 4; ++nt) {
        v2f B;                                       // z in B-frag layout
        B.x = zt[buf][ja][nt * 16 + m];
        B.y = zt[buf][ja + 1][nt * 16 + m];
        acc[nt] = __builtin_amdgcn_wmma_f32_16x16x4_f32(
            false, A, false, B, (short)0, acc[nt], false, false);
      }
    }
    asyncWait0();
    __syncthreads();
  }

  // denom: lanes l and l+16 handled the same row m -> one xor-16 add
  const float dtot = dpart + __shfl_xor(dpart, 16);
  const size_t rowbase = (size_t)blockIdx.y * NN + (size_t)i0;
  if (kh == 0) denomP[rowbase + m] = dtot;
  #pragma unroll
  for (int nt = 0; nt < 4; ++nt)
    #pragma unroll
    for (int v = 0; v < 8; ++v)
      accP[(rowbase + (size_t)((kh << 3) + v)) * FOUT + nt * 16 + m] = acc[nt][v];
}

// ------------------------------------------------------------------
// Kernel E: out = (sum_c accP) / (sum_c denomP) + bias
// ------------------------------------------------------------------
__global__ __launch_bounds__(256) void finalize_kernel(
    const float* __restrict__ accP, const float* __restrict__ denomP,
    const float* __restrict__ bias, float* __restrict__ out)
{
  const int idx = blockIdx.x * 256 + threadIdx.x;   // 0 .. 8192*64-1
  const int i = idx >> 6;
  const int n = idx & 63;
  float d = 0.f, s = 0.f;
  #pragma unroll
  for (int c = 0; c < 4; ++c) {
    d += denomP[(size_t)c * NN + i];
    s += accP[((size_t)c * NN + i) * FOUT + n];
  }
  out[idx] = s / d + bias[n];
}

// ------------------------------------------------------------------
extern "C" void kernel_launch(void* const* d_in, const int* in_sizes, int n_in,
                              void* d_out, int out_size, void* d_ws, size_t ws_size,
                              hipStream_t stream)
{
  (void)in_sizes; (void)n_in; (void)out_size; (void)ws_size;

  const float* input = (const float*)d_in[0];     // (8192, 512) f32
  const int*   adj   = (const int*)  d_in[1];     // (8192, 8192) i32
  const float* W     = (const float*)d_in[2];     // (512, 64) f32
  const float* a     = (const float*)d_in[3];     // (128, 1) f32
  const float* bias  = (const float*)d_in[4];     // (64,) f32
  float*       out   = (float*)d_out;             // (8192, 64) f32

  float* ws     = (float*)d_ws;
  float* zbuf   = ws;                             // 8192*64
  float* s1     = zbuf + (size_t)NN * FOUT;       // 8192
  float* s2     = s1 + NN;                        // 8192
  float* s2m    = s2 + NN;                        // 1 (+pad)
  float* accP   = s2m + 64;                       // 4 * 8192*64
  float* denomP = accP + (size_t)4 * NN * FOUT;   // 4 * 8192

  zgemm_kernel   <<<NN / 16,      32, 0, stream>>>(input, W, zbuf);
  s12_kernel     <<<NN / 8,      256, 0, stream>>>(zbuf, a, s1, s2);
  s2max_kernel   <<<1,           256, 0, stream>>>(s2, s2m);
  attn_kernel    <<<dim3(NN/128, 4), 256, 0, stream>>>(adj, zbuf, s1, s2, s2m, accP, denomP);
  finalize_kernel<<<(NN * FOUT) / 256, 256, 0, stream>>>(accP, denomP, bias, out);
}